// DifferentiableMPO_cvx_70497593197464
// MI455X (gfx1250) — compile-verified
//
#include <hip/hip_runtime.h>
#include <math.h>

// Problem constants from the reference
#define BB 128
#define HH 12
#define NN 128
#define CCOEF 0.01f      // C = cost_coeff * 10 = 0.01
#define NITERS 300
#define BISECT 30

typedef float v2f __attribute__((ext_vector_type(2)));
typedef float v8f __attribute__((ext_vector_type(8)));

__device__ __forceinline__ float wave_sum(float v) {
#pragma unroll
  for (int m = 16; m >= 1; m >>= 1) v += __shfl_xor(v, m, 32);
  return v;
}
__device__ __forceinline__ float wave_max(float v) {
#pragma unroll
  for (int m = 16; m >= 1; m >>= 1) v = fmaxf(v, __shfl_xor(v, m, 32));
  return v;
}

// ---------------------------------------------------------------------------
// Phase A: Sigma_{b,h} = L L^T  (128x128x128, fp32) via V_WMMA_F32_16X16X4_F32
// Also emits fro[bh] = ||Sigma||_F.  One block per (b,h), 8 waves.
// Wave w computes the 16-row stripe r0 = 16*w, looping over 8 column tiles.
// A fragment (16x4 f32): lane m=lane&15, k-pair kq=lane>>4 -> float2 {k,k+1}.
// B fragment (4x16 f32): same pattern with lane holding the N index; since
// B = L^T, both fragments are float2 row reads of L.
// ---------------------------------------------------------------------------
__global__ __launch_bounds__(256) void sigma_kernel(const float* __restrict__ L,
                                                    float* __restrict__ Sigma,
                                                    float* __restrict__ fro) {
  const int bh   = blockIdx.x;
  const int tid  = threadIdx.x;
  const int wave = tid >> 5;
  const int lane = tid & 31;
  const int m    = lane & 15;   // M index for A, N index for B
  const int kq   = lane >> 4;   // which K pair (0 -> K0,1 ; 1 -> K2,3)

  const float* Lb = L     + (size_t)bh * (NN * NN);
  float*       Sb = Sigma + (size_t)bh * (NN * NN);

  const int r0 = wave * 16;
  float fr = 0.f;
  __shared__ float fpart[8];

  const float* Arow = Lb + (r0 + m) * NN + 2 * kq;

  for (int c0 = 0; c0 < NN; c0 += 16) {
    const float* Brow = Lb + (c0 + m) * NN + 2 * kq;
    v8f acc = {};
#pragma unroll 4
    for (int kk = 0; kk < NN; kk += 4) {
      v2f a = *(const v2f*)(Arow + kk);
      v2f b = *(const v2f*)(Brow + kk);
      // D = A*B + C   (args: neg_a, A, neg_b, B, c_mod, C, reuse_a, reuse_b)
      acc = __builtin_amdgcn_wmma_f32_16x16x4_f32(false, a, false, b,
                                                  (short)0, acc, false, false);
    }
    // C/D layout: VGPR r -> (M = r + 8*kq, N = m)
#pragma unroll
    for (int r = 0; r < 8; ++r) {
      const int row = r0 + r + 8 * kq;
      const int col = c0 + m;
      Sb[row * NN + col] = acc[r];
      fr += acc[r] * acc[r];
    }
  }

  fr = wave_sum(fr);
  if (lane == 0) fpart[wave] = fr;
  __syncthreads();
  if (tid == 0) {
    float t = 0.f;
#pragma unroll
    for (int w = 0; w < 8; ++w) t += fpart[w];
    fro[bh] = sqrtf(t);
  }
}

// ---------------------------------------------------------------------------
// Phase B: FISTA. One block per batch b, 12 waves; wave h owns head h.
// y double-buffered in LDS; Sigma streamed (L2-resident) with the symmetric
// trick so reads are coalesced float4 rows; simplex projection by bisection
// done entirely inside each wave32.
// ---------------------------------------------------------------------------
__global__ __launch_bounds__(384) void fista_kernel(const float* __restrict__ Sigma,
                                                    const float* __restrict__ fro,
                                                    const float* __restrict__ wprev,
                                                    float* __restrict__ out) {
  const int b    = blockIdx.x;
  const int tid  = threadIdx.x;
  const int h    = tid >> 5;
  const int lane = tid & 31;
  const int i0   = lane * 4;

  __shared__ float Y[2][HH][NN];
  __shared__ float WP[NN];

  // step = 1 / (2*max_h fro + 8C), computed redundantly per wave
  float f = (lane < HH) ? fro[b * HH + lane] : 0.f;
  const float step = 1.f / (2.f * wave_max(f) + 8.f * CCOEF);

  // init: W = Y = broadcast(w_prev)
  float4 wp = *(const float4*)(wprev + b * NN + i0);
  if (h == 0) *(float4*)&WP[i0] = wp;
  float w0 = wp.x, w1 = wp.y, w2 = wp.z, w3 = wp.w;   // W for own head
  *(float4*)&Y[0][h][i0] = wp;
  __syncthreads();

  const float* Sh = Sigma + ((size_t)b * HH + h) * (NN * NN);
  float tk = 1.f;

  for (int it = 0; it < NITERS; ++it) {
    const int cur = it & 1, nxt = cur ^ 1;

    // t = Sigma @ y  using symmetry: t_i = sum_j Sigma[j][i] * y_j
    float a0 = 0.f, a1 = 0.f, a2 = 0.f, a3 = 0.f;
#pragma unroll 2
    for (int j0 = 0; j0 < NN; j0 += 4) {
      float4 y4 = *(const float4*)&Y[cur][h][j0];
      const float* r = Sh + j0 * NN + i0;
      float4 s0 = *(const float4*)(r);
      float4 s1 = *(const float4*)(r + NN);
      float4 s2 = *(const float4*)(r + 2 * NN);
      float4 s3 = *(const float4*)(r + 3 * NN);
      __builtin_prefetch(r + 8 * NN, 0, 0);   // global_prefetch_b8 ahead of stream
      a0 += s0.x * y4.x + s1.x * y4.y + s2.x * y4.z + s3.x * y4.w;
      a1 += s0.y * y4.x + s1.y * y4.y + s2.y * y4.z + s3.y * y4.w;
      a2 += s0.z * y4.x + s1.z * y4.y + s2.z * y4.z + s3.z * y4.w;
      a3 += s0.w * y4.x + s1.w * y4.y + s2.w * y4.z + s3.w * y4.w;
    }

    // gradient with temporal chain coupling
    float4 yc = *(const float4*)&Y[cur][h][i0];
    float4 ym = (h == 0) ? *(const float4*)&WP[i0]
                         : *(const float4*)&Y[cur][h - 1][i0];
    float dn0 = 0.f, dn1 = 0.f, dn2 = 0.f, dn3 = 0.f;
    if (h < HH - 1) {
      float4 yp = *(const float4*)&Y[cur][h + 1][i0];
      dn0 = yp.x - yc.x; dn1 = yp.y - yc.y; dn2 = yp.z - yc.z; dn3 = yp.w - yc.w;
    }
    const float twoC = 2.f * CCOEF;
    float g0 = 2.f * a0 + twoC * ((yc.x - ym.x) - dn0);
    float g1 = 2.f * a1 + twoC * ((yc.y - ym.y) - dn1);
    float g2 = 2.f * a2 + twoC * ((yc.z - ym.z) - dn2);
    float g3 = 2.f * a3 + twoC * ((yc.w - ym.w) - dn3);

    float v0 = yc.x - step * g0;
    float v1 = yc.y - step * g1;
    float v2 = yc.z - step * g2;
    float v3 = yc.w - step * g3;

    // simplex projection by bisection on theta in [vmax-1, vmax]
    float vm = wave_max(fmaxf(fmaxf(v0, v1), fmaxf(v2, v3)));
    float lo = vm - 1.f, hi = vm;
#pragma unroll 1
    for (int q = 0; q < BISECT; ++q) {
      float th = 0.5f * (lo + hi);
      float s = fmaxf(v0 - th, 0.f) + fmaxf(v1 - th, 0.f) +
                fmaxf(v2 - th, 0.f) + fmaxf(v3 - th, 0.f);
      s = wave_sum(s);
      if (s >= 1.f) lo = th; else hi = th;
    }
    const float th = 0.5f * (lo + hi);
    float n0 = fmaxf(v0 - th, 0.f);
    float n1 = fmaxf(v1 - th, 0.f);
    float n2 = fmaxf(v2 - th, 0.f);
    float n3 = fmaxf(v3 - th, 0.f);

    // momentum (scalar sequence, uniform across lanes)
    const float tn   = 0.5f * (1.f + sqrtf(1.f + 4.f * tk * tk));
    const float beta = (tk - 1.f) / tn;
    tk = tn;

    float4 yn;
    yn.x = n0 + beta * (n0 - w0);
    yn.y = n1 + beta * (n1 - w1);
    yn.z = n2 + beta * (n2 - w2);
    yn.w = n3 + beta * (n3 - w3);
    w0 = n0; w1 = n1; w2 = n2; w3 = n3;

    *(float4*)&Y[nxt][h][i0] = yn;
    __syncthreads();   // publishes Y[nxt] for neighbor reads next iteration
  }

  float4 wf; wf.x = w0; wf.y = w1; wf.z = w2; wf.w = w3;
  *(float4*)(out + ((size_t)b * HH + h) * NN + i0) = wf;
}

extern "C" void kernel_launch(void* const* d_in, const int* in_sizes, int n_in,
                              void* d_out, int out_size, void* d_ws, size_t ws_size,
                              hipStream_t stream) {
  (void)in_sizes; (void)n_in; (void)out_size; (void)ws_size;
  // inputs: d_in[0] = mu (unused), d_in[1] = L (B,H,N,N) f32, d_in[2] = w_prev (B,N) f32
  const float* L     = (const float*)d_in[1];
  const float* wprev = (const float*)d_in[2];

  float* fro   = (float*)d_ws;        // 1536 floats (padded to 2048)
  float* Sigma = fro + 2048;          // 128*12*128*128 floats = 96 MB (L2-resident)

  sigma_kernel<<<BB * HH, 256, 0, stream>>>(L, Sigma, fro);
  fista_kernel<<<BB, 384, 0, stream>>>(Sigma, fro, wprev, (float*)d_out);
}